// TarMAC_90280212562559
// MI455X (gfx1250) — compile-verified
//
#include <hip/hip_runtime.h>

#define D_H   128
#define D_SIG 16

typedef __attribute__((ext_vector_type(2))) float v2f;
typedef __attribute__((ext_vector_type(8))) float v8f;

// Accumulate a 16x16 fp32 tile: acc += A[16xK] * B[Kx16] using V_WMMA_F32_16X16X4_F32.
// A row-major with leading dim lda, B row-major with leading dim ldb (tile at B[.., col0] passed in).
// ISA VGPR layout (32-bit A 16x4): lanes 0-15 -> M=lane, K={k,k+1}; lanes 16-31 -> M=lane-16, K={k+2,k+3}.
// B 4x16 mirrored; C/D: VGPR r holds (M=r, lanes 0-15) and (M=r+8, lanes 16-31), N=lane&15.
__device__ __forceinline__ v8f wmma_tile_f32(const float* __restrict__ A, int lda,
                                             const float* __restrict__ B, int ldb,
                                             int K, v8f acc, int lane) {
  const int half = lane >> 4;
  const int l16  = lane & 15;
  const float* arow = A + (size_t)l16 * lda + 2 * half;
  const float* bcol = B + l16 + (size_t)(2 * half) * ldb;
#pragma unroll 4
  for (int k = 0; k < K; k += 4) {
    v2f a, b;
    a.x = arow[k];
    a.y = arow[k + 1];
    b.x = bcol[(size_t)k * ldb];
    b.y = bcol[(size_t)(k + 1) * ldb];
    acc = __builtin_amdgcn_wmma_f32_16x16x4_f32(false, a, false, b, (short)0, acc, false, false);
  }
  return acc;
}

// out[row0:row0+16, :Nout] = (relu?)(A[row0:row0+16, :K] @ W[K,Nout] + bias)
// one wave per 16x16 output tile; blockDim.x/32 waves cover Nout columns.
__global__ void gemm_bias_kernel(const float* __restrict__ A, const float* __restrict__ W,
                                 const float* __restrict__ bias, float* __restrict__ out,
                                 int K, int Nout, int relu) {
  const int wave = threadIdx.x >> 5;
  const int lane = threadIdx.x & 31;
  const int row0 = blockIdx.x * 16;
  const int col0 = wave * 16;
  if (col0 >= Nout) return;
  v8f acc = {};
  acc = wmma_tile_f32(A + (size_t)row0 * K, K, W + col0, Nout, K, acc, lane);
  const int half = lane >> 4, l16 = lane & 15;
  const float b = bias[col0 + l16];
#pragma unroll
  for (int r = 0; r < 8; ++r) {
    float v = acc[r] + b;
    if (relu) v = fmaxf(v, 0.0f);
    out[(size_t)(row0 + r + 8 * half) * Nout + col0 + l16] = v;
  }
}

// inv_norm[i] = 1 / max(||sigraw[i,:16]||, 1e-12)
__global__ void invnorm_kernel(const float* __restrict__ sigraw, float* __restrict__ invn, int n) {
  const int i = blockIdx.x * blockDim.x + threadIdx.x;
  if (i >= n) return;
  const float4* p = (const float4*)(sigraw + (size_t)i * D_SIG);
  float s = 0.0f;
#pragma unroll
  for (int j = 0; j < 4; ++j) {
    float4 v = p[j];
    s += v.x * v.x + v.y * v.y + v.z * v.z + v.w * v.w;
  }
  invn[i] = 1.0f / fmaxf(sqrtf(s), 1e-12f);
}

// per edge: score = <sig[src], sig[dst]> (normalized); e = exp(score); denom[src] += e
// scores are in [-1,1] (normalized sigs), so softmax max-shift is unnecessary for stability.
__global__ void edge_score_kernel(const int* __restrict__ ei, const float* __restrict__ sigraw,
                                  const float* __restrict__ invn, float* __restrict__ e_arr,
                                  float* __restrict__ denom, int E) {
  const int e = blockIdx.x * blockDim.x + threadIdx.x;
  if (e >= E) return;
  const int s = ei[e];
  const int d = ei[E + e];
  const float4* ps = (const float4*)(sigraw + (size_t)s * D_SIG);
  const float4* pd = (const float4*)(sigraw + (size_t)d * D_SIG);
  float dot = 0.0f;
#pragma unroll
  for (int j = 0; j < 4; ++j) {
    float4 a = ps[j], b = pd[j];
    dot += a.x * b.x + a.y * b.y + a.z * b.z + a.w * b.w;
  }
  const float score = dot * invn[s] * invn[d];
  const float ex = expf(score);
  e_arr[e] = ex;
  atomicAdd(denom + s, ex);
}

// one wave per edge: comm[src, :] += (e/denom[src]) * msg[dst, :]  (float4 per lane)
__global__ void scatter_kernel(const int* __restrict__ ei, const float* __restrict__ e_arr,
                               const float* __restrict__ denom, const float* __restrict__ msg,
                               float* __restrict__ comm, int E) {
  const int edge = (int)((blockIdx.x * (size_t)blockDim.x + threadIdx.x) >> 5);
  const int lane = threadIdx.x & 31;
  if (edge >= E) return;
  const int s = ei[edge];
  const int d = ei[E + edge];
  const float w = e_arr[edge] / denom[s];
  const float4 v = ((const float4*)(msg + (size_t)d * D_H))[lane];
  float* pc = comm + (size_t)s * D_H + lane * 4;
  atomicAdd(pc + 0, w * v.x);
  atomicAdd(pc + 1, w * v.y);
  atomicAdd(pc + 2, w * v.z);
  atomicAdd(pc + 3, w * v.w);
}

// combined = relu([h | comm] @ agg_w + agg_b) staged in LDS; out = combined @ dec_w + dec_b
#define LDS_STRIDE 132
__global__ void agg_dec_kernel(const float* __restrict__ h, const float* __restrict__ comm,
                               const float* __restrict__ agg_w, const float* __restrict__ agg_b,
                               const float* __restrict__ dec_w, const float* __restrict__ dec_b,
                               float* __restrict__ out) {
  __shared__ float lds[16 * LDS_STRIDE];
  const int wave = threadIdx.x >> 5;
  const int lane = threadIdx.x & 31;
  const int half = lane >> 4, l16 = lane & 15;
  const int row0 = blockIdx.x * 16;
  const int col0 = wave * 16;

  v8f acc = {};
  acc = wmma_tile_f32(h + (size_t)row0 * D_H, D_H, agg_w + col0, D_H, D_H, acc, lane);
  acc = wmma_tile_f32(comm + (size_t)row0 * D_H, D_H, agg_w + (size_t)D_H * D_H + col0, D_H, D_H, acc, lane);
  const float b1 = agg_b[col0 + l16];
#pragma unroll
  for (int r = 0; r < 8; ++r)
    lds[(r + 8 * half) * LDS_STRIDE + col0 + l16] = fmaxf(acc[r] + b1, 0.0f);

  __syncthreads();

  v8f acc2 = {};
  acc2 = wmma_tile_f32(lds, LDS_STRIDE, dec_w + col0, D_H, D_H, acc2, lane);
  const float b2 = dec_b[col0 + l16];
#pragma unroll
  for (int r = 0; r < 8; ++r)
    out[(size_t)(row0 + r + 8 * half) * D_H + col0 + l16] = acc2[r] + b2;
}

static inline size_t align16(size_t x) { return (x + 15) & ~(size_t)15; }

extern "C" void kernel_launch(void* const* d_in, const int* in_sizes, int n_in,
                              void* d_out, int out_size, void* d_ws, size_t ws_size,
                              hipStream_t stream) {
  const float* x     = (const float*)d_in[0];
  const int*   ei    = (const int*)d_in[1];
  const float* enc_w = (const float*)d_in[2];
  const float* enc_b = (const float*)d_in[3];
  const float* sig_w = (const float*)d_in[4];
  const float* sig_b = (const float*)d_in[5];
  const float* msg_w = (const float*)d_in[6];
  const float* msg_b = (const float*)d_in[7];
  const float* agg_w = (const float*)d_in[8];
  const float* agg_b = (const float*)d_in[9];
  const float* dec_w = (const float*)d_in[10];
  const float* dec_b = (const float*)d_in[11];
  float* out = (float*)d_out;

  const int n = in_sizes[0] / D_H;   // 50000 (multiple of 16)
  const int E = in_sizes[1] / 2;     // 800000

  // workspace carve-up (all fp32)
  char* base = (char*)d_ws;
  size_t off = 0;
  float* h      = (float*)(base + off); off = align16(off + (size_t)n * D_H * 4);
  float* msg    = (float*)(base + off); off = align16(off + (size_t)n * D_H * 4);
  float* comm   = (float*)(base + off); off = align16(off + (size_t)n * D_H * 4);
  float* sigraw = (float*)(base + off); off = align16(off + (size_t)n * D_SIG * 4);
  float* invn   = (float*)(base + off); off = align16(off + (size_t)n * 4);
  float* e_arr  = (float*)(base + off); off = align16(off + (size_t)E * 4);
  float* denom  = (float*)(base + off); off = align16(off + (size_t)n * 4);
  (void)ws_size; (void)n_in; (void)out_size;

  // zero the accumulators every call (graph-capture safe)
  hipMemsetAsync(comm, 0, (size_t)n * D_H * 4, stream);
  hipMemsetAsync(denom, 0, (size_t)n * 4, stream);

  const int row_tiles = n / 16;

  // h = x @ enc_w + enc_b
  gemm_bias_kernel<<<row_tiles, 256, 0, stream>>>(x, enc_w, enc_b, h, D_H, D_H, 0);
  // sigraw = h @ sig_w + sig_b   (single 16-col tile -> one wave)
  gemm_bias_kernel<<<row_tiles, 32, 0, stream>>>(h, sig_w, sig_b, sigraw, D_H, D_SIG, 0);
  // msg = relu(h @ msg_w + msg_b)
  gemm_bias_kernel<<<row_tiles, 256, 0, stream>>>(h, msg_w, msg_b, msg, D_H, D_H, 1);
  // inv norms
  invnorm_kernel<<<(n + 255) / 256, 256, 0, stream>>>(sigraw, invn, n);
  // per-edge exp(score) + denom
  edge_score_kernel<<<(E + 255) / 256, 256, 0, stream>>>(ei, sigraw, invn, e_arr, denom, E);
  // weighted scatter of messages (one wave per edge)
  scatter_kernel<<<(E * 32 + 255) / 256, 256, 0, stream>>>(ei, e_arr, denom, msg, comm, E);
  // combined = relu([h|comm] @ agg_w + agg_b); out = combined @ dec_w + dec_b
  agg_dec_kernel<<<row_tiles, 256, 0, stream>>>(h, comm, agg_w, agg_b, dec_w, dec_b, out);
}